// NeuralVMEmbedding_46548855554106
// MI455X (gfx1250) — compile-verified
//
#include <hip/hip_runtime.h>
#include <hip/hip_bf16.h>
#include <cstdint>

// NeuralVMEmbedding for MI455X (gfx1250, wave32).
// Store-bandwidth-bound embedding gather + masked one-hot fixups.
//   - async global->LDS staging of the token row (ASYNCcnt path)
//   - non-temporal b128 stores for the 512MB write stream
//   - wave-uniform metadata in SGPRs (readfirstlane), saddr-form gathers
//   - self-sufficient blocks (prefix cummax carry computed from LDS copy)

#define CODE_START 256
#define CODE_END   257
#define MEM_TOK    258
#define ADDR_KEY   206
#define MEM_STORE  455
#define DMODEL     512
#define SEQ        8192
#define BROWS      32
#define POS_PER_BLK 128
#define THREADS     256

typedef float v4f __attribute__((ext_vector_type(4)));

__global__ __launch_bounds__(THREADS) void
neural_vm_embed_kernel(const int* __restrict__ tok,
                       const float* __restrict__ emb,
                       const int* __restrict__ mhe_ptr,
                       float* __restrict__ out,
                       int S) {
    __shared__ int s_tok[SEQ];          // whole token row (32 KB)
    __shared__ int s_mtok[POS_PER_BLK]; // per-position token
    __shared__ int s_maf[POS_PER_BLK];  // addr[15:0] | mask<<16 | mem<<17
    __shared__ int s_carry, s_fce;

    const int tid = threadIdx.x;
    const int b   = blockIdx.y;                 // batch row (no division)
    const int p0  = blockIdx.x * POS_PER_BLK;   // first position of this block
    const int* rowTok = tok + (size_t)b * S;

    // ---- Phase A1: async DMA of the token row into LDS (CDNA5 async path) ----
    // GVS addressing: SGPR 64-bit base + per-lane 32-bit byte offset in VGPR.
    {
        uint32_t lds_base = (uint32_t)(uintptr_t)(void*)s_tok; // low 32 bits = LDS offset
        const int n16 = S >> 2; // 16-byte chunks (2048)
        for (int i = tid; i < n16; i += THREADS) {
            uint32_t loff = lds_base + (uint32_t)(i << 4);
            uint32_t goff = (uint32_t)(i << 4);
            asm volatile("global_load_async_to_lds_b128 %0, %1, %2"
                         :: "v"(loff), "v"(goff), "s"(rowTok)
                         : "memory");
        }
        asm volatile("s_wait_asynccnt 0" ::: "memory");
    }
    __syncthreads();

    // ---- Phase A2: carry-in cummax over [0,p0) and row-wide first CODE_END ----
    int myCS = -1;   // max pos of CODE_START in [0, p0)
    int myCE = S;    // min pos of CODE_END in [0, S)
    for (int s = tid; s < S; s += THREADS) {
        int t = s_tok[s];
        if (t == CODE_END && s < myCE) myCE = s;
        if (t == CODE_START && s < p0 && s > myCS) myCS = s;
    }
    if (tid == 0) { s_carry = -1; s_fce = S; }
    __syncthreads();
    atomicMax(&s_carry, myCS);
    atomicMin(&s_fce,  myCE);
    __syncthreads();

    // ---- Phase A3: per-position metadata (tiny serial cummax over 128) ----
    if (tid == 0) {
        int cs  = s_carry;
        int fce = s_fce;
        int mhe = *mhe_ptr;
        for (int k = 0; k < POS_PER_BLK; ++k) {
            int p = p0 + k;
            int t = s_tok[p];
            if (t == CODE_START) cs = p;
            int seq = p - cs - 1; if (seq < 0) seq = 0;
            int addr = (seq / 5) * 8 + seq % 5;        // < 16384, fits 16 bits
            int m  = (cs >= 0 && p < fce && t < 256) ? 1 : 0;
            int mm = (t == MEM_TOK && p < mhe) ? 1 : 0;
            s_mtok[k] = t;
            s_maf[k]  = (addr & 0xFFFF) | (m << 16) | (mm << 17);
        }
    }
    __syncthreads();

    // ---- Phase B: gather + fixups + non-temporal b128 stores ----
    // Thread layout: c4 = tid & 127 is a FIXED float4 column group per thread;
    // k = 2*i + (tid>>7) is WAVE-UNIFORM -> metadata lives in SGPRs.
    const int h   = tid >> 7;      // 0 or 1, wave-uniform
    const int c4  = tid & 127;     // float4 column group, fixed per thread
    const int col = c4 << 2;
    const bool inA = (c4 >= (ADDR_KEY >> 2)) && (c4 <= ((ADDR_KEY + 47) >> 2)); // 51..63
    const bool inM = (c4 == (MEM_STORE >> 2));                                  // 113

    const float* embc = emb + (size_t)col;                                 // + t*512 per pos
    float* outc = out + ((size_t)b * S + (size_t)p0) * DMODEL + (size_t)col;

#pragma unroll 4
    for (int i = 0; i < POS_PER_BLK / 2; ++i) {
        int k  = (i << 1) + h;
        int t  = __builtin_amdgcn_readfirstlane(s_mtok[k]);  // scalar token
        int af = __builtin_amdgcn_readfirstlane(s_maf[k]);   // scalar addr|flags

        v4f v = *(const v4f*)(embc + (size_t)t * DMODEL);    // saddr-form gather

        if (inA && (af & 0x10000)) {       // one-hot nibbles into cols [206,254)
            int addr = af & 0xFFFF;
            int lo  = addr & 15;
            int hi  = (addr >> 4) & 15;
            int top = (addr >> 8) & 15;
#pragma unroll
            for (int e = 0; e < 4; ++e) {
                int j = col + e - ADDR_KEY;                  // loop-invariant per thread
                bool hit = ((unsigned)j < 16u) ? (lo == j)
                         : ((unsigned)j < 32u) ? (hi == j - 16)
                         : ((unsigned)j < 48u) ? (top == j - 32)
                                               : false;
                if (hit) v[e] = 1.0f;
            }
        }
        if (inM && (af & 0x20000)) v.w = 1.0f;               // col 455 (MEM_STORE)

        __builtin_nontemporal_store(v, (v4f*)(outc + (size_t)k * DMODEL));
    }
}

extern "C" void kernel_launch(void* const* d_in, const int* in_sizes, int n_in,
                              void* d_out, int out_size, void* d_ws, size_t ws_size,
                              hipStream_t stream) {
    const int*   tok = (const int*)d_in[0];     // (32, 8192) int32
    const float* emb = (const float*)d_in[1];   // (272, 512) f32
    const int*   mhe = (const int*)d_in[2];     // scalar int
    float*       out = (float*)d_out;           // (32, 8192, 512) f32

    int S = in_sizes[0] / BROWS;                // 8192
    int chunks = S / POS_PER_BLK;               // 64 chunks per row
    dim3 grid(chunks, BROWS);                   // 2D grid: no integer division
    dim3 block(THREADS);                        // 8 wave32s
    hipLaunchKernelGGL(neural_vm_embed_kernel, grid, block, 0, stream,
                       tok, emb, mhe, out, S);
}